// LSTM_Cell_63848983822341
// MI455X (gfx1250) — compile-verified
//
#include <hip/hip_runtime.h>

typedef __attribute__((ext_vector_type(2))) float v2f;
typedef __attribute__((ext_vector_type(8))) float v8f;

#define BATCH   262144
#define INSZ    128
#define H       16
#define KTOT    144      // INSZ + H
#define KSTEPS  36       // KTOT / 4
#define NTILES  (BATCH / 16)
#define NBLOCKS 1024
#define TPB     256

__device__ __forceinline__ float fast_sigmoid(float v) {
    return 1.0f / (1.0f + __expf(-v));
}
__device__ __forceinline__ float fast_tanh(float v) {
    // tanh(v) = 2*sigmoid(2v) - 1  (saturates correctly at +/-1)
    return 2.0f / (1.0f + __expf(-2.0f * v)) - 1.0f;
}

// 8-byte NT global load that stays a single global_load_b64
__device__ __forceinline__ v2f nt_load_v2f(const float* p) {
    unsigned long long raw =
        __builtin_nontemporal_load((const unsigned long long*)p);
    return __builtin_bit_cast(v2f, raw);
}

__global__ __launch_bounds__(TPB) void lstm_cell_wmma(
    const float* __restrict__ x,      const float* __restrict__ h_prev,
    const float* __restrict__ c_prev,
    const float* __restrict__ Wf, const float* __restrict__ bf,
    const float* __restrict__ Wi, const float* __restrict__ bi,
    const float* __restrict__ Wc, const float* __restrict__ bc,
    const float* __restrict__ Wo, const float* __restrict__ bo,
    float* __restrict__ out_h, float* __restrict__ out_c)
{
    // wT[g][k][n] = W_g[n][k]  (transposed weights, 4*144*16 f32 = 36 KB)
    __shared__ float wT[4 * KTOT * H];

    const int tid = threadIdx.x;

    // ---- stage all 4 gate weight matrices, transposed, into LDS ----
    for (int idx = tid; idx < 4 * KTOT * H; idx += TPB) {
        int g = idx / (KTOT * H);
        int r = idx - g * (KTOT * H);
        int k = r / H;
        int n = r - k * H;
        const float* Wg = (g == 0) ? Wf : (g == 1) ? Wi : (g == 2) ? Wc : Wo;
        wT[idx] = Wg[n * KTOT + k];
    }
    __syncthreads();

    const int lane = tid & 31;
    const int m    = lane & 15;   // batch row within tile (A) / hidden unit n (B,D)
    const int kh   = lane >> 4;   // 0 -> K={0,1}, 1 -> K={2,3} within a k-step

    const int wave   = (tid >> 5) + blockIdx.x * (TPB / 32);
    const int nwaves = NBLOCKS * (TPB / 32);

    // per-lane biases (hidden index = m)
    const float vbf = bf[m], vbi = bi[m], vbc = bc[m], vbo = bo[m];

    // ---- B fragments are tile-invariant: load ALL of them once per wave ----
    // bfrag[g][t]: VGPR0 = K=0/2 row, VGPR1 = K=1/3 row of the 4x16 W^T slice
    v2f bfrag[4 * KSTEPS];
    #pragma unroll
    for (int t = 0; t < KSTEPS; ++t) {
        const int kb = 4 * t + 2 * kh;
        #pragma unroll
        for (int g = 0; g < 4; ++g) {
            v2f b;
            b.x = wT[(g * KTOT + kb    ) * H + m];
            b.y = wT[(g * KTOT + kb + 1) * H + m];
            bfrag[g * KSTEPS + t] = b;
        }
    }

    #pragma unroll 1
    for (int tile = wave; tile < NTILES; tile += nwaves) {
        const int row0 = tile * 16;
        const int arow = row0 + m;

        // ---- A fragments: 36 b64 NT loads, saddr + 32-bit voffset + imm ----
        v2f afrag[KSTEPS];
        {
            const int xbase = arow * INSZ + 2 * kh;   // element index, fits in i32
            #pragma unroll
            for (int t = 0; t < 32; ++t)
                afrag[t] = nt_load_v2f(x + xbase + 4 * t);
            const int hbase = arow * H + 2 * kh;
            #pragma unroll
            for (int t = 0; t < 4; ++t)
                afrag[32 + t] = nt_load_v2f(h_prev + hbase + 4 * t);
        }

        v8f accf = {}, acci = {}, accc = {}, acco = {};

        #pragma unroll
        for (int t = 0; t < KSTEPS; ++t) {
            const v2f a = afrag[t];
            accf = __builtin_amdgcn_wmma_f32_16x16x4_f32(
                false, a, false, bfrag[0 * KSTEPS + t], (short)0, accf, false, false);
            acci = __builtin_amdgcn_wmma_f32_16x16x4_f32(
                false, a, false, bfrag[1 * KSTEPS + t], (short)0, acci, false, false);
            accc = __builtin_amdgcn_wmma_f32_16x16x4_f32(
                false, a, false, bfrag[2 * KSTEPS + t], (short)0, accc, false, false);
            acco = __builtin_amdgcn_wmma_f32_16x16x4_f32(
                false, a, false, bfrag[3 * KSTEPS + t], (short)0, acco, false, false);
        }

        // ---- epilogue: D layout -> VGPR j holds row row0 + j + 8*kh, col m ----
        const int obase = (row0 + 8 * kh) * H + m;    // + j*H per accumulator slot
        #pragma unroll
        for (int j = 0; j < 8; ++j) {
            const int off = obase + j * H;

            const float fgate = fast_sigmoid(accf[j] + vbf);
            const float igate = fast_sigmoid(acci[j] + vbi);
            const float ctil  = fast_tanh   (accc[j] + vbc);
            const float ogate = fast_sigmoid(acco[j] + vbo);

            const float cp = __builtin_nontemporal_load(c_prev + off);
            const float cn = fgate * cp + igate * ctil;
            const float hn = ogate * fast_tanh(cn);

            __builtin_nontemporal_store(hn, out_h + off);
            __builtin_nontemporal_store(cn, out_c + off);
        }
    }
}

extern "C" void kernel_launch(void* const* d_in, const int* in_sizes, int n_in,
                              void* d_out, int out_size, void* d_ws, size_t ws_size,
                              hipStream_t stream) {
    const float* x      = (const float*)d_in[0];
    const float* h_prev = (const float*)d_in[1];
    const float* c_prev = (const float*)d_in[2];
    const float* Wf     = (const float*)d_in[3];
    const float* bf     = (const float*)d_in[4];
    const float* Wi     = (const float*)d_in[5];
    const float* bi     = (const float*)d_in[6];
    const float* Wc     = (const float*)d_in[7];
    const float* bc     = (const float*)d_in[8];
    const float* Wo     = (const float*)d_in[9];
    const float* bo     = (const float*)d_in[10];

    float* out_h = (float*)d_out;                    // tuple order: (h_t, C_t)
    float* out_c = out_h + (size_t)BATCH * H;

    lstm_cell_wmma<<<NBLOCKS, TPB, 0, stream>>>(
        x, h_prev, c_prev, Wf, bf, Wi, bi, Wc, bc, Wo, bo, out_h, out_c);
}